// MetaConvSmoother_4509715660935
// MI455X (gfx1250) — compile-verified
//
#include <hip/hip_runtime.h>
#include <hip/hip_bf16.h>
#include <cstdint>

#define BATCH 64
#define NPIX  512
#define MLCH  3

typedef float v2f  __attribute__((ext_vector_type(2)));
typedef float v8f  __attribute__((ext_vector_type(8)));
typedef unsigned int u32x4 __attribute__((ext_vector_type(4)));
typedef int   i32x4 __attribute__((ext_vector_type(4)));
typedef int   i32x8 __attribute__((ext_vector_type(8)));

#if defined(__has_builtin)
#  if __has_builtin(__builtin_amdgcn_tensor_load_to_lds)
#    define HAVE_TDM 1
#  endif
#endif
#ifndef HAVE_TDM
#  define HAVE_TDM 0
#endif

__device__ __forceinline__ float gelu_exact(float x) {
    return 0.5f * x * (1.0f + erff(x * 0.7071067811865476f));
}

// ---------------------------------------------------------------------------
// Kernel 1: hypernetwork. One block per MLP (grid.x == 2), 28 waves.
// GEMM1: H[64,100] = gelu(V[64,9] @ W1^T + b1)   (K padded 9->12, 3 WMMA steps)
// GEMM2: O[64,147] = H @ W2^T + b2               (K=100, 25 WMMA steps)
// Both on V_WMMA_F32_16X16X4_F32 (fp32). Zero-padding is done with clamped
// addresses + multiplicative masks so every load is unconditional (no EXEC
// branching around loads).
// Output layout: wAll[m*294 + sel*147 + (ml*49 + i*7 + j)]
// ---------------------------------------------------------------------------
__global__ __launch_bounds__(896)
void hyper_wmma_kernel(const float* __restrict__ kernelA,
                       const float* __restrict__ fc1_w1, const float* __restrict__ fc1_b1,
                       const float* __restrict__ fc1_w2, const float* __restrict__ fc1_b2,
                       const float* __restrict__ fc2_w1, const float* __restrict__ fc2_b1,
                       const float* __restrict__ fc2_w2, const float* __restrict__ fc2_b2,
                       float* __restrict__ wAll) {
    const int sel = blockIdx.x;
    const float* w1 = sel ? fc2_w1 : fc1_w1;
    const float* b1 = sel ? fc2_b1 : fc1_b1;
    const float* w2 = sel ? fc2_w2 : fc1_w2;
    const float* b2 = sel ? fc2_b2 : fc1_b2;

    __shared__ float Hs[64 * 112];   // H, N padded 100 -> 112 (zero-filled pad)

    const int tid   = threadIdx.x;
    const int wid   = tid >> 5;
    const int lane  = tid & 31;
    const int mrow  = lane & 15;          // A-matrix row within tile
    const int kgrp  = (lane >> 4) << 1;   // K pair base: 0 or 2
    const int ncol  = lane & 15;          // B/C column within tile
    const int mhalf = (lane >> 4) << 3;   // C rows: +0 or +8

    // ---- GEMM1 ----  28 waves == 4 (M) x 7 (N) tiles exactly
    {
        const int mt = wid / 7;
        const int nt = wid % 7;
        const int m  = mt * 16 + mrow;                 // < 64 always
        const int n16 = nt * 16 + ncol;                // < 112
        const float nmask = (n16 < 100) ? 1.0f : 0.0f;
        const int   nc    = (n16 < 100) ? n16 : 0;     // clamped (valid) address
        v8f c = {};
        #pragma unroll
        for (int ks = 0; ks < 3; ++ks) {
            const int k0 = ks * 4 + kgrp;
            const int ka = (k0     < 9) ? k0     : 0;
            const int kb = (k0 + 1 < 9) ? k0 + 1 : 0;
            const float ma = (k0     < 9) ? 1.0f : 0.0f;
            const float mb = (k0 + 1 < 9) ? 1.0f : 0.0f;
            v2f a, bb;
            a.x  = kernelA[m * 9 + ka] * ma;
            a.y  = kernelA[m * 9 + kb] * mb;
            bb.x = w1[nc * 9 + ka] * (ma * nmask);
            bb.y = w1[nc * 9 + kb] * (mb * nmask);
            c = __builtin_amdgcn_wmma_f32_16x16x4_f32(false, a, false, bb,
                                                      (short)0, c, false, false);
        }
        const float bias = b1[nc];
        #pragma unroll
        for (int v = 0; v < 8; ++v) {
            const int mm = mt * 16 + mhalf + v;
            const float g = gelu_exact(c[v] + bias);
            Hs[mm * 112 + n16] = (n16 < 100) ? g : 0.0f;
        }
    }
    __syncthreads();

    // ---- GEMM2 ----  4 (M) x 10 (N) = 40 tiles over 28 waves
    for (int t = wid; t < 40; t += 28) {
        const int mt = t / 10;
        const int nt = t % 10;
        const int n16 = nt * 16 + ncol;
        const float nmask = (n16 < 147) ? 1.0f : 0.0f;
        const int   nc    = (n16 < 147) ? n16 : 0;
        const int   m     = mt * 16 + mrow;
        v8f c = {};
        for (int ks = 0; ks < 25; ++ks) {         // K = 100 exactly
            const int k0 = ks * 4 + kgrp;         // even, <= 98
            v2f a = *(const v2f*)&Hs[m * 112 + k0];   // one ds_load_b64
            v2f bb;
            bb.x = w2[nc * 100 + k0]     * nmask;
            bb.y = w2[nc * 100 + k0 + 1] * nmask;
            c = __builtin_amdgcn_wmma_f32_16x16x4_f32(false, a, false, bb,
                                                      (short)0, c, false, false);
        }
        if (n16 < 147) {
            const float bias = b2[n16];
            #pragma unroll
            for (int v = 0; v < 8; ++v) {
                const int mm = mt * 16 + mhalf + v;
                wAll[(size_t)mm * 294 + sel * 147 + n16] = c[v] + bias;
            }
        }
    }
}

// ---------------------------------------------------------------------------
// Kernel 2: fused  r = f - Ax ;  tmp = conv7(r) ;  out = x + conv7(tmp)
// One 32x32 output tile per workgroup, 256 threads, all staging in LDS.
// Conv stages use 2x2 output micro-tiles with an 8-wide register row window
// (4x ds_load_b64 per tap-row feeding 196 FMAs per channel-tile).
// ---------------------------------------------------------------------------
#define TW  32
#define XE  46           // TW + 14 (x halo: 6 conv + 1 for 3x3)
#define RE  44           // TW + 12
#define TE  38           // TW + 6
#define SMX 48           // row stride for Xs / rs (even)
#define SMT 40           // row stride for tmp     (even)
#define OFF_W   0        // 294 generated weights (w1: [0,147), w2: [147,294))
#define OFF_KA  294      // 9 floats of kernelA for this sample
#define OFF_X   304      // 46*48 = 2208
#define OFF_R   2512     // 44*48 = 2112
#define OFF_T   4624     // 3*38*40 = 4560
#define SM_TOT  9184     // floats = 36736 bytes

__global__ __launch_bounds__(256)
void metaconv_kernel(const float* __restrict__ x, const float* __restrict__ f,
                     const float* __restrict__ kernelA,
                     const float* __restrict__ wAll,
                     float* __restrict__ out) {
    __shared__ float sm[SM_TOT];
    const int tid = threadIdx.x;
    const int b   = blockIdx.z;
    const int ty0 = blockIdx.y * TW;
    const int tx0 = blockIdx.x * TW;

#if HAVE_TDM
    // TDM: DMA the 294 per-sample generated weights (1 x 294 f32 tile) to LDS 0.
    if (tid < 32) {
        const unsigned long long ga =
            (unsigned long long)(const void*)(wAll + (size_t)b * 294);
        u32x4 g0;
        g0[0] = 1u;                                   // count=1, user mode
        g0[1] = (unsigned)(OFF_W * 4);                // lds_addr (bytes)
        g0[2] = (unsigned)(ga & 0xFFFFFFFFull);       // global_addr lo
        g0[3] = ((unsigned)(ga >> 32) & 0x01FFFFFFu) | (2u << 30);  // hi | type=2
        i32x8 g1;
        g1[0] = 0x00020000;   // workgroup_mask=0, data_size=2 (4B)
        g1[1] = 0x01260000;   // tensor_dim0[15:0] = 294
        g1[2] = 0x00010000;   // tensor_dim1 = 1
        g1[3] = 0x01260000;   // tile_dim0 = 294
        g1[4] = 0x00000001;   // tile_dim1 = 1, tile_dim2 = 0
        g1[5] = 294;          // tensor_dim0_stride
        g1[6] = 0x01260000;   // tensor_dim1_stride = 294
        g1[7] = 0;
        i32x4 g2 = {0, 0, 0, 0};
        i32x4 g3 = {0, 0, 0, 0};
#if __clang_major__ >= 23
        i32x8 g4 = {0, 0, 0, 0, 0, 0, 0, 0};
        __builtin_amdgcn_tensor_load_to_lds(g0, g1, g2, g3, g4, 0);
#else
        __builtin_amdgcn_tensor_load_to_lds(g0, g1, g2, g3, 0);
#endif
        __builtin_amdgcn_s_wait_tensorcnt(0);
    }
#endif

    if (tid < 9) sm[OFF_KA + tid] = kernelA[b * 9 + tid];

    // Stage X window [ty0-7, ty0+39) x [tx0-7, tx0+39), asymmetric pad:
    // bottom/right out-of-range -> 1 (takes precedence), top/left -> 0.
    for (int i = tid; i < XE * XE; i += 256) {
        const int r = i / XE, c = i % XE;
        const int gy = ty0 - 7 + r, gx = tx0 - 7 + c;
        float v;
        if (gy >= NPIX || gx >= NPIX) v = 1.0f;
        else if (gy < 0 || gx < 0)    v = 0.0f;
        else v = x[((size_t)b * NPIX + gy) * NPIX + gx];
        sm[OFF_X + r * SMX + c] = v;
    }
    __syncthreads();

    // Redundant direct weight load: guarantees correctness regardless of the
    // (hardware-unverified) TDM descriptor semantics; writes identical values.
    for (int i = tid; i < 294; i += 256)
        sm[OFF_W + i] = wAll[(size_t)b * 294 + i];

    // r = f - Ax on [ty0-6, ty0+38)^2 ; zero outside the image (conv1 zero-pad)
    for (int i = tid; i < RE * RE; i += 256) {
        const int r = i / RE, c = i % RE;
        const int gy = ty0 - 6 + r, gx = tx0 - 6 + c;
        float v = 0.0f;
        if (gy >= 0 && gy < NPIX && gx >= 0 && gx < NPIX) {
            float ax = 0.0f;
            #pragma unroll
            for (int di = 0; di < 3; ++di)
                #pragma unroll
                for (int dj = 0; dj < 3; ++dj)
                    ax = fmaf(sm[OFF_KA + di * 3 + dj],
                              sm[OFF_X + (r + di) * SMX + (c + dj)], ax);
            v = f[((size_t)b * NPIX + gy) * NPIX + gx] - ax;
        }
        sm[OFF_R + r * SMX + c] = v;
    }
    __syncthreads();

    // conv1: tmp[ml] on [ty0-3, ty0+35)^2, 2x2 micro-tiles (19x19=361 tiles).
    // Zero outside image: conv2 sees a zero-padded [N,N] feature map.
    for (int ml = 0; ml < MLCH; ++ml) {
        float w[49];
        #pragma unroll
        for (int t = 0; t < 49; ++t) w[t] = sm[OFF_W + ml * 49 + t];
        for (int t2 = tid; t2 < 19 * 19; t2 += 256) {
            const int tr = (t2 / 19) * 2;        // even row in [0,36]
            const int tc = (t2 % 19) * 2;        // even col in [0,36]
            float a00 = 0.0f, a01 = 0.0f, a10 = 0.0f, a11 = 0.0f;
            #pragma unroll
            for (int ky = 0; ky < 8; ++ky) {
                float row[8];
                const float* rp = &sm[OFF_R + (tr + ky) * SMX + tc];  // 8B aligned
                #pragma unroll
                for (int j = 0; j < 4; ++j) {
                    const v2f t = ((const v2f*)rp)[j];                 // ds_load_b64
                    row[2 * j] = t.x; row[2 * j + 1] = t.y;
                }
                if (ky < 7) {                    // top output row, ki = ky
                    #pragma unroll
                    for (int kj = 0; kj < 7; ++kj) {
                        a00 = fmaf(w[ky * 7 + kj], row[kj],     a00);
                        a01 = fmaf(w[ky * 7 + kj], row[kj + 1], a01);
                    }
                }
                if (ky >= 1) {                   // bottom output row, ki = ky-1
                    #pragma unroll
                    for (int kj = 0; kj < 7; ++kj) {
                        a10 = fmaf(w[(ky - 1) * 7 + kj], row[kj],     a10);
                        a11 = fmaf(w[(ky - 1) * 7 + kj], row[kj + 1], a11);
                    }
                }
            }
            #pragma unroll
            for (int py = 0; py < 2; ++py)
                #pragma unroll
                for (int px = 0; px < 2; ++px) {
                    const int r = tr + py, c = tc + px;
                    const int gy = ty0 - 3 + r, gx = tx0 - 3 + c;
                    const bool in = (gy >= 0 && gy < NPIX && gx >= 0 && gx < NPIX);
                    const float a = (py == 0) ? (px == 0 ? a00 : a01)
                                              : (px == 0 ? a10 : a11);
                    sm[OFF_T + (ml * TE + r) * SMT + c] = in ? a : 0.0f;
                }
        }
    }
    __syncthreads();

    // conv2 + residual: exactly one 2x2 output micro-tile per thread.
    const int tr = (tid >> 4) << 1;   // even row in [0,30]
    const int tc = (tid & 15) << 1;   // even col in [0,30]
    float a00 = 0.0f, a01 = 0.0f, a10 = 0.0f, a11 = 0.0f;
    for (int ml = 0; ml < MLCH; ++ml) {
        float w[49];
        #pragma unroll
        for (int t = 0; t < 49; ++t) w[t] = sm[OFF_W + 147 + ml * 49 + t];
        const int base = OFF_T + ml * TE * SMT;
        #pragma unroll
        for (int ky = 0; ky < 8; ++ky) {
            float row[8];
            const float* rp = &sm[base + (tr + ky) * SMT + tc];       // 8B aligned
            #pragma unroll
            for (int j = 0; j < 4; ++j) {
                const v2f t = ((const v2f*)rp)[j];                     // ds_load_b64
                row[2 * j] = t.x; row[2 * j + 1] = t.y;
            }
            if (ky < 7) {
                #pragma unroll
                for (int kj = 0; kj < 7; ++kj) {
                    a00 = fmaf(w[ky * 7 + kj], row[kj],     a00);
                    a01 = fmaf(w[ky * 7 + kj], row[kj + 1], a01);
                }
            }
            if (ky >= 1) {
                #pragma unroll
                for (int kj = 0; kj < 7; ++kj) {
                    a10 = fmaf(w[(ky - 1) * 7 + kj], row[kj],     a10);
                    a11 = fmaf(w[(ky - 1) * 7 + kj], row[kj + 1], a11);
                }
            }
        }
    }
    // Paired (float2) residual load + store: 8B aligned since tx0+tc is even.
    {
        const size_t g0 = ((size_t)b * NPIX + (ty0 + tr)) * NPIX + (tx0 + tc);
        const size_t g1 = ((size_t)b * NPIX + (ty0 + tr + 1)) * NPIX + (tx0 + tc);
        const v2f x0 = *(const v2f*)&x[g0];
        const v2f x1 = *(const v2f*)&x[g1];
        v2f o0, o1;
        o0.x = x0.x + a00; o0.y = x0.y + a01;
        o1.x = x1.x + a10; o1.y = x1.y + a11;
        *(v2f*)&out[g0] = o0;
        *(v2f*)&out[g1] = o1;
    }
}

// ---------------------------------------------------------------------------
extern "C" void kernel_launch(void* const* d_in, const int* in_sizes, int n_in,
                              void* d_out, int out_size, void* d_ws, size_t ws_size,
                              hipStream_t stream) {
    const float* x      = (const float*)d_in[0];
    const float* f      = (const float*)d_in[1];
    const float* kA     = (const float*)d_in[2];
    const float* fc1_w1 = (const float*)d_in[3];
    const float* fc1_b1 = (const float*)d_in[4];
    const float* fc1_w2 = (const float*)d_in[5];
    const float* fc1_b2 = (const float*)d_in[6];
    const float* fc2_w1 = (const float*)d_in[7];
    const float* fc2_b1 = (const float*)d_in[8];
    const float* fc2_w2 = (const float*)d_in[9];
    const float* fc2_b2 = (const float*)d_in[10];
    float* wAll = (float*)d_ws;          // 64 * 294 floats of generated weights
    float* out  = (float*)d_out;

    hipLaunchKernelGGL(hyper_wmma_kernel, dim3(2), dim3(896), 0, stream,
                       kA, fc1_w1, fc1_b1, fc1_w2, fc1_b2,
                       fc2_w1, fc2_b1, fc2_w2, fc2_b2, wAll);

    hipLaunchKernelGGL(metaconv_kernel,
                       dim3(NPIX / TW, NPIX / TW, BATCH), dim3(256), 0, stream,
                       x, f, kA, wAll, out);
}